// _GraphConvolution_10934986735661
// MI455X (gfx1250) — compile-verified
//
#include <hip/hip_runtime.h>
#include <hip/hip_bf16.h>

typedef __attribute__((ext_vector_type(2))) float v2f;
typedef __attribute__((ext_vector_type(8))) float v8f;
typedef __attribute__((ext_vector_type(4))) int   v4i;

#define ALPHA 0.1f
#define Nn   64
#define Dd   2000
#define Bb   512
#define OUTD 256
#define KC   16

#if __has_builtin(__builtin_amdgcn_global_load_async_to_lds_b128) && \
    __has_builtin(__builtin_amdgcn_s_wait_asynccnt)
#define HAS_ASYNC_LDS 1
#else
#define HAS_ASYNC_LDS 0
#endif

#if HAS_ASYNC_LDS
// Builtin signature (from hipcc diagnostic): (v4i AS1*, v4i AS3*, imm offset, imm cpol)
#define GPTR(p) ((__attribute__((address_space(1))) v4i*)(p))
#define LPTR(p) ((__attribute__((address_space(3))) v4i*)(p))
#define ASYNC_B128(g, l) __builtin_amdgcn_global_load_async_to_lds_b128(GPTR(g), LPTR(l), 0, 0)
#endif

// ---------------------------------------------------------------------------
// Kernel 1: nv = tanh(ALPHA * (emb @ W + b))   [64,2000] @ [2000,2000]
// ---------------------------------------------------------------------------
__global__ __launch_bounds__(256) void nv_kernel(const float* __restrict__ emb,
                                                 const float* __restrict__ W,
                                                 const float* __restrict__ bias,
                                                 float* __restrict__ out) {
  __shared__ float es[8][256];
  const int t  = threadIdx.x;
  const int d  = blockIdx.x * 256 + t;
  const int i0 = blockIdx.y * 8;
  const bool act = (d < Dd);
  float acc[8];
  const float bv = act ? bias[d] : 0.f;
#pragma unroll
  for (int r = 0; r < 8; ++r) acc[r] = bv;
  for (int k0 = 0; k0 < Dd; k0 += 256) {
    const int kc = min(256, Dd - k0);
#pragma unroll
    for (int r = 0; r < 8; ++r)
      if (t < kc) es[r][t] = emb[(i0 + r) * Dd + k0 + t];
    __syncthreads();
    if (act) {
      for (int k = 0; k < kc; ++k) {
        const float w = W[(k0 + k) * Dd + d];
#pragma unroll
        for (int r = 0; r < 8; ++r) acc[r] += es[r][k] * w;
      }
    }
    __syncthreads();
  }
  if (act) {
#pragma unroll
    for (int r = 0; r < 8; ++r)
      out[(i0 + r) * Dd + d] = tanhf(ALPHA * acc[r]);
  }
}

// ---------------------------------------------------------------------------
// Kernel 2: a[i][j] = nv1[i].nv2[j] - nv2[i].nv1[j]   (64 blocks, row each)
// ---------------------------------------------------------------------------
__global__ __launch_bounds__(256) void a_kernel(const float* __restrict__ nv1,
                                                const float* __restrict__ nv2,
                                                float* __restrict__ amat) {
  __shared__ float red[256];
  const int i   = blockIdx.x;
  const int t   = threadIdx.x;
  const int j   = t >> 2;
  const int sub = t & 3;
  const float* r1 = nv1 + i * Dd;
  const float* r2 = nv2 + i * Dd;
  const float* c1 = nv1 + j * Dd;
  const float* c2 = nv2 + j * Dd;
  float s = 0.f;
  for (int n = 0; n < Dd / 4; ++n) {
    const int d = sub * (Dd / 4) + n;
    s += r1[d] * c2[d] - r2[d] * c1[d];
  }
  red[t] = s;
  __syncthreads();
  if (sub == 0)
    amat[i * 64 + j] = red[j * 4] + red[j * 4 + 1] + red[j * 4 + 2] + red[j * 4 + 3];
}

// ---------------------------------------------------------------------------
// Kernel 3 (one block): adj = relu(tanh(ALPHA*a)); per-row top-3 mask;
// ahat = adj*mask + I; anorm = Dinv * ahat^T * Dinv; a3 = anorm^3.
// ---------------------------------------------------------------------------
__global__ __launch_bounds__(256) void graph_kernel(const float* __restrict__ amat,
                                                    float* __restrict__ a3out) {
  __shared__ float adj[64][64];   // reused as a2 later
  __shared__ float ah[64][64];
  __shared__ float an[64][64];
  __shared__ float dinv[64];
  const int t = threadIdx.x;
  for (int e = t; e < 4096; e += 256) {
    const float v = tanhf(ALPHA * amat[e]);
    adj[e >> 6][e & 63] = v > 0.f ? v : 0.f;
    ah[e >> 6][e & 63]  = 0.f;
  }
  __syncthreads();
  if (t < 64) {
    // top-3 with jax tie-break (largest value, lowest index first)
    int c0 = -1, c1 = -1, c2 = -1;
    for (int p = 0; p < 3; ++p) {
      float best = -1.f; int bi = 0;
      for (int jj = 0; jj < 64; ++jj) {
        if (jj == c0 || jj == c1 || jj == c2) continue;
        const float v = adj[t][jj];
        if (v > best) { best = v; bi = jj; }
      }
      if (p == 0) c0 = bi; else if (p == 1) c1 = bi; else c2 = bi;
      ah[t][bi] = adj[t][bi];
    }
    ah[t][t] += 1.f;                 // self loop
  }
  __syncthreads();
  if (t < 64) {                      // in-degree = column sums
    float s = 0.f;
    for (int ii = 0; ii < 64; ++ii) s += ah[ii][t];
    dinv[t] = s > 0.f ? rsqrtf(s) : 0.f;
  }
  __syncthreads();
  for (int e = t; e < 4096; e += 256) {
    const int ii = e >> 6, jj = e & 63;
    an[ii][jj] = dinv[ii] * ah[jj][ii] * dinv[jj];   // Dinv * ahat^T * Dinv
  }
  __syncthreads();
  for (int e = t; e < 4096; e += 256) {              // a2 = an @ an (into adj)
    const int ii = e >> 6, jj = e & 63;
    float s = 0.f;
    for (int k = 0; k < 64; ++k) s += an[ii][k] * an[k][jj];
    adj[ii][jj] = s;
  }
  __syncthreads();
  for (int e = t; e < 4096; e += 256) {              // a3 = a2 @ an
    const int ii = e >> 6, jj = e & 63;
    float s = 0.f;
    for (int k = 0; k < 64; ++k) s += adj[ii][k] * an[k][jj];
    a3out[e] = s;
  }
}

// ---------------------------------------------------------------------------
// Kernel 4: Y[(b,n),o] = X[(b,n),:] @ sg_w[:,o]   M=32768, K=2000, N=256.
// WMMA f32 16x16x4. Workgroup tile 64x64, 8 waves as 4(M) x 2(N).
// Double-buffered LDS staging via GLOBAL_LOAD_ASYNC_TO_LDS_B128 (ASYNCcnt).
// A-tile row stride = 20 floats: 80B (16B-aligned for b128) and 20*r mod 64
// distinct for r=0..15 -> conflict-free 16-rows-same-column fragment reads.
// ---------------------------------------------------------------------------
__global__ __launch_bounds__(256) void proj_kernel(const float* __restrict__ X,
                                                   const float* __restrict__ W,
                                                   float* __restrict__ Y) {
  __shared__ __align__(16) float As[2][64][20];
  __shared__ __align__(16) float Bs[2][KC][64];
  const int t    = threadIdx.x;
  const int m0   = blockIdx.x * 64;
  const int n0   = blockIdx.y * 64;
  const int wave = t >> 5;
  const int lane = t & 31;
  const int wm   = wave & 3;         // 0..3 -> 16-row strip
  const int wn   = wave >> 2;        // 0..1 -> 32-col strip
  const int lrow = lane & 15;
  const int lhi  = lane >> 4;        // lanes 16-31 carry K=2,3 (A) / M+8 (C)
  const int arow = t >> 2, aq = t & 3;    // A tile: 64 rows x 16 k, 1 b128/thread
  const int brow = t >> 4, bq = t & 15;   // B tile: 16 k  x 64 n, 1 b128/thread
  v8f acc0 = {0.f, 0.f, 0.f, 0.f, 0.f, 0.f, 0.f, 0.f};
  v8f acc1 = {0.f, 0.f, 0.f, 0.f, 0.f, 0.f, 0.f, 0.f};

  const float* gA = &X[(size_t)(m0 + arow) * Dd + aq * 4];        // + k0
  const float* gB = &W[(size_t)brow * OUTD + n0 + bq * 4];        // + k0*OUTD

#if HAS_ASYNC_LDS
  // ---- async double-buffered pipeline ----
  ASYNC_B128(gA, &As[0][arow][aq * 4]);
  ASYNC_B128(gB, &Bs[0][brow][bq * 4]);
  int buf = 0;
  for (int k0 = 0; k0 < Dd; k0 += KC) {
    const bool more = (k0 + KC) < Dd;
    if (more) {
      ASYNC_B128(gA + (k0 + KC), &As[buf ^ 1][arow][aq * 4]);
      ASYNC_B128(gB + (size_t)(k0 + KC) * OUTD, &Bs[buf ^ 1][brow][bq * 4]);
      __builtin_amdgcn_s_wait_asynccnt(2);   // retire this buffer's 2 copies
    } else {
      __builtin_amdgcn_s_wait_asynccnt(0);
    }
    __syncthreads();
#pragma unroll
    for (int kk = 0; kk < KC; kk += 4) {
      v2f a;
      a.x = As[buf][wm * 16 + lrow][kk + 2 * lhi];
      a.y = As[buf][wm * 16 + lrow][kk + 2 * lhi + 1];
      v2f b0, b1;
      b0.x = Bs[buf][kk + 2 * lhi][wn * 32 + lrow];
      b0.y = Bs[buf][kk + 2 * lhi + 1][wn * 32 + lrow];
      b1.x = Bs[buf][kk + 2 * lhi][wn * 32 + 16 + lrow];
      b1.y = Bs[buf][kk + 2 * lhi + 1][wn * 32 + 16 + lrow];
      acc0 = __builtin_amdgcn_wmma_f32_16x16x4_f32(false, a, false, b0,
                                                   (short)0, acc0, false, false);
      acc1 = __builtin_amdgcn_wmma_f32_16x16x4_f32(false, a, false, b1,
                                                   (short)0, acc1, false, false);
    }
    __syncthreads();
    buf ^= 1;
  }
#else
  // ---- fallback: register-pipelined synchronous staging ----
  float4 ra = *(const float4*)gA;
  float4 rb = *(const float4*)gB;
  for (int k0 = 0; k0 < Dd; k0 += KC) {
    As[0][arow][aq * 4 + 0] = ra.x;
    As[0][arow][aq * 4 + 1] = ra.y;
    As[0][arow][aq * 4 + 2] = ra.z;
    As[0][arow][aq * 4 + 3] = ra.w;
    *(float4*)&Bs[0][brow][bq * 4] = rb;
    __syncthreads();
    if (k0 + KC < Dd) {   // issue next chunk's loads before compute
      ra = *(const float4*)(gA + (k0 + KC));
      rb = *(const float4*)(gB + (size_t)(k0 + KC) * OUTD);
    }
#pragma unroll
    for (int kk = 0; kk < KC; kk += 4) {
      v2f a;
      a.x = As[0][wm * 16 + lrow][kk + 2 * lhi];
      a.y = As[0][wm * 16 + lrow][kk + 2 * lhi + 1];
      v2f b0, b1;
      b0.x = Bs[0][kk + 2 * lhi][wn * 32 + lrow];
      b0.y = Bs[0][kk + 2 * lhi + 1][wn * 32 + lrow];
      b1.x = Bs[0][kk + 2 * lhi][wn * 32 + 16 + lrow];
      b1.y = Bs[0][kk + 2 * lhi + 1][wn * 32 + 16 + lrow];
      acc0 = __builtin_amdgcn_wmma_f32_16x16x4_f32(false, a, false, b0,
                                                   (short)0, acc0, false, false);
      acc1 = __builtin_amdgcn_wmma_f32_16x16x4_f32(false, a, false, b1,
                                                   (short)0, acc1, false, false);
    }
    __syncthreads();
  }
#endif

#pragma unroll
  for (int r = 0; r < 8; ++r) {      // C layout: vgpr r, lanes16-31 -> M = r+8
    const int row = m0 + wm * 16 + lhi * 8 + r;
    const int col = n0 + wn * 32 + lrow;
    Y[(size_t)row * OUTD + col]      = acc0[r];
    Y[(size_t)row * OUTD + col + 16] = acc1[r];
  }
}

// ---------------------------------------------------------------------------
// Kernel 5: out[b] = relu(A3 @ Y[b] + bias), in place in d_out.
// Whole 64x64 tile staged in LDS (async) before any write -> in-place safe.
// ---------------------------------------------------------------------------
__global__ __launch_bounds__(256) void apply_kernel(const float* __restrict__ A3,
                                                    const float* __restrict__ sgb,
                                                    float* __restrict__ Y) {
  __shared__ __align__(16) float As[64][68];   // 272B row stride: aligned + conflict-free
  __shared__ __align__(16) float Bs[64][64];
  const int t  = threadIdx.x;
  const int b  = blockIdx.x;
  const int n0 = blockIdx.y * 64;
  const float* gY = &Y[(size_t)b * (Nn * OUTD) + n0];

#if HAS_ASYNC_LDS
#pragma unroll
  for (int i = 0; i < 4; ++i) {                // 1024 float4 slots / 256 threads
    const int e4  = t + i * 256;
    const int row = e4 >> 4;                   // 64 rows
    const int c   = (e4 & 15) * 4;             // 16 b128 segs per row
    ASYNC_B128(&A3[row * 64 + c], &As[row][c]);
    ASYNC_B128(gY + (size_t)row * OUTD + c, &Bs[row][c]);
  }
  __builtin_amdgcn_s_wait_asynccnt(0);
#else
#pragma unroll
  for (int i = 0; i < 4; ++i) {
    const int e4  = t + i * 256;
    const int row = e4 >> 4;
    const int c   = (e4 & 15) * 4;
    const float4 va = *(const float4*)&A3[row * 64 + c];
    As[row][c + 0] = va.x; As[row][c + 1] = va.y;
    As[row][c + 2] = va.z; As[row][c + 3] = va.w;
    *(float4*)&Bs[row][c] = *(const float4*)(gY + (size_t)row * OUTD + c);
  }
#endif
  __syncthreads();

  const int wave = t >> 5, lane = t & 31;
  const int wm = wave & 3, wn = wave >> 2;
  const int lrow = lane & 15, lhi = lane >> 4;
  v8f acc0 = {0.f, 0.f, 0.f, 0.f, 0.f, 0.f, 0.f, 0.f};
  v8f acc1 = {0.f, 0.f, 0.f, 0.f, 0.f, 0.f, 0.f, 0.f};
#pragma unroll
  for (int kk = 0; kk < 64; kk += 4) {
    v2f a;
    a.x = As[wm * 16 + lrow][kk + 2 * lhi];
    a.y = As[wm * 16 + lrow][kk + 2 * lhi + 1];
    v2f b0, b1;
    b0.x = Bs[kk + 2 * lhi][wn * 32 + lrow];
    b0.y = Bs[kk + 2 * lhi + 1][wn * 32 + lrow];
    b1.x = Bs[kk + 2 * lhi][wn * 32 + 16 + lrow];
    b1.y = Bs[kk + 2 * lhi + 1][wn * 32 + 16 + lrow];
    acc0 = __builtin_amdgcn_wmma_f32_16x16x4_f32(false, a, false, b0,
                                                 (short)0, acc0, false, false);
    acc1 = __builtin_amdgcn_wmma_f32_16x16x4_f32(false, a, false, b1,
                                                 (short)0, acc1, false, false);
  }
#pragma unroll
  for (int r = 0; r < 8; ++r) {
    const int row = wm * 16 + lhi * 8 + r;
    const int c0  = n0 + wn * 32 + lrow;
    const float v0 = acc0[r] + sgb[c0];
    const float v1 = acc1[r] + sgb[c0 + 16];
    Y[(size_t)b * (Nn * OUTD) + row * OUTD + c0]      = v0 > 0.f ? v0 : 0.f;
    Y[(size_t)b * (Nn * OUTD) + row * OUTD + c0 + 16] = v1 > 0.f ? v1 : 0.f;
  }
}

// ---------------------------------------------------------------------------
extern "C" void kernel_launch(void* const* d_in, const int* in_sizes, int n_in,
                              void* d_out, int out_size, void* d_ws, size_t ws_size,
                              hipStream_t stream) {
  const float* x    = (const float*)d_in[0];
  const float* emb1 = (const float*)d_in[1];
  const float* emb2 = (const float*)d_in[2];
  const float* l1w  = (const float*)d_in[3];
  const float* l1b  = (const float*)d_in[4];
  const float* l2w  = (const float*)d_in[5];
  const float* l2b  = (const float*)d_in[6];
  const float* sgw  = (const float*)d_in[7];
  const float* sgb  = (const float*)d_in[8];
  float* out = (float*)d_out;
  float* ws  = (float*)d_ws;

  float* nv1  = ws;                 // 64*2000
  float* nv2  = ws + 128000;        // 64*2000
  float* amat = ws + 256000;        // 64*64
  float* a3   = ws + 260096;        // 64*64

  nv_kernel<<<dim3(8, 8), 256, 0, stream>>>(emb1, l1w, l1b, nv1);
  nv_kernel<<<dim3(8, 8), 256, 0, stream>>>(emb2, l2w, l2b, nv2);
  a_kernel<<<dim3(64), 256, 0, stream>>>(nv1, nv2, amat);
  graph_kernel<<<dim3(1), 256, 0, stream>>>(amat, a3);
  // Y = X @ sg_w  (pre-activation), straight into d_out
  proj_kernel<<<dim3((Bb * Nn) / 64, OUTD / 64), 256, 0, stream>>>(x, sgw, out);
  // out = relu(A3 @ Y + b), in place
  apply_kernel<<<dim3(Bb, OUTD / 64), 256, 0, stream>>>(a3, sgb, out);
}